// Arch7V4Layer_33243046871800
// MI455X (gfx1250) — compile-verified
//
#include <hip/hip_runtime.h>
#include <hip/hip_bf16.h>

#define HDIM 128
#define NT_CAP 100000
#define BN_ROWS 256
#define WP 288  // paired-W LDS row stride (floats); 288 % 64 == 32 -> disjoint bank halves

typedef __attribute__((ext_vector_type(2))) float v2f;
typedef __attribute__((ext_vector_type(8))) float v8f;

// ---------------------------------------------------------------------------
// Zero fill
// ---------------------------------------------------------------------------
__global__ void zero_f32(float* p, long long n) {
  long long i = (long long)blockIdx.x * blockDim.x + threadIdx.x;
  long long stride = (long long)gridDim.x * blockDim.x;
  for (; i < n; i += stride) p[i] = 0.0f;
}

// ---------------------------------------------------------------------------
// Edge message: agg[dst] += relu(x[src] + ea[e]); one thread per (edge, 4 floats)
// ---------------------------------------------------------------------------
__global__ void edge_msg(const float* __restrict__ x, const int* __restrict__ ei,
                         const float* __restrict__ ea, float* agg, long long E) {
  long long t = (long long)blockIdx.x * blockDim.x + threadIdx.x;
  if (t >= E * 32) return;
  long long e = t >> 5;
  int c = (int)(t & 31) << 2;
  long long src = ei[e];
  long long dst = ei[E + e];
  const float4 xv = *(const float4*)(x + src * HDIM + c);
  const float4 av = *(const float4*)(ea + e * HDIM + c);
  float4 m;
  m.x = fmaxf(xv.x + av.x, 0.0f);
  m.y = fmaxf(xv.y + av.y, 0.0f);
  m.z = fmaxf(xv.z + av.z, 0.0f);
  m.w = fmaxf(xv.w + av.w, 0.0f);
  float* p = agg + dst * HDIM + c;
  unsafeAtomicAdd(p + 0, m.x);
  unsafeAtomicAdd(p + 1, m.y);
  unsafeAtomicAdd(p + 2, m.z);
  unsafeAtomicAdd(p + 3, m.w);
}

// ---------------------------------------------------------------------------
// Scatter-sum rows (for scatter_mean): sum[id] += x[src]; cnt[id] += 1
// Optional gather indirection: srcRow = gather ? gather[row] : row.
// Mask: ids[srcRow] < 0 skipped (equivalent to reference's weight-0 rows).
// ---------------------------------------------------------------------------
__global__ void scatter_rows(const float* __restrict__ x, const int* __restrict__ ids,
                             const int* __restrict__ gather,
                             float* sum, float* cnt, long long N) {
  long long t = (long long)blockIdx.x * blockDim.x + threadIdx.x;
  if (t >= N * 32) return;
  long long row = t >> 5;
  int c = (int)(t & 31) << 2;
  long long srcRow = gather ? (long long)gather[row] : row;
  int id = ids[srcRow];
  if (id < 0) return;
  const float4 v = *(const float4*)(x + srcRow * HDIM + c);
  float* p = sum + (long long)id * HDIM + c;
  unsafeAtomicAdd(p + 0, v.x);
  unsafeAtomicAdd(p + 1, v.y);
  unsafeAtomicAdd(p + 2, v.z);
  unsafeAtomicAdd(p + 3, v.w);
  if (c == 0) unsafeAtomicAdd(&cnt[id], 1.0f);
}

// ---------------------------------------------------------------------------
// sum[r] /= max(cnt[r], 1) for r < min(Ncap, *dN)
// ---------------------------------------------------------------------------
__global__ void div_rows(float* sum, const float* __restrict__ cnt,
                         int Ncap, const int* __restrict__ dN) {
  int Neff = Ncap;
  if (dN) { int m = *dN; Neff = m < Ncap ? m : Ncap; }
  long long t = (long long)blockIdx.x * blockDim.x + threadIdx.x;
  if (t >= (long long)Ncap * 32) return;
  long long r = t >> 5;
  if (r >= Neff) return;
  int c = (int)(t & 31) << 2;
  float w = 1.0f / fmaxf(cnt[r], 1.0f);
  float4 v = *(float4*)(sum + r * HDIM + c);
  v.x *= w; v.y *= w; v.z *= w; v.w *= w;
  *(float4*)(sum + r * HDIM + c) = v;
}

// ---------------------------------------------------------------------------
// WMMA GEMM: Out[M x 128] = epi( A_eff[M x 128] @ W[128 x 128] + bias )
//   A_eff row g = A[rowIdx ? rowIdx[g] : g] (+ A2[g] if A2)
//   epi = relu if doRelu
// Block = 128 threads (4 wave32), 64 output rows per block.
//
// LDS layouts chosen so every WMMA fragment is a single 8B-aligned b64 load
// into an even-aligned VGPR pair (no repack movs):
//   sW: pair-packed, sW[p*WP + n*2 + q] = W[2p+q][n]  (b fragment contiguous)
//       WP=288 -> lanes 0-15 (pair p) / 16-31 (pair p+1) use disjoint banks.
//   sA: row-major, stride 130 floats (520B, 8B aligned; k0 even).
//
// Fragment layouts per CDNA5 ISA 7.12.2:
//   A 16x4 f32: lanes 0-15 = M, vgpr{0,1} = K{0,1}; lanes 16-31 = K{2,3}
//   B 4x16 f32: lanes 0-15 = N with K{0,1}; lanes 16-31 = K{2,3}
//   C/D 16x16:  vgpr r -> M = r + 8*(lane>=16), N = lane%16
// ---------------------------------------------------------------------------
__global__ __launch_bounds__(128) void gemm128_wmma(
    const float* A, const float* A2, const int* rowIdx,
    const float* W, const float* bias,
    float* Out, int Mcap, const int* dM, int doRelu) {
  __shared__ float sW[64 * WP];   // 73728 B
  __shared__ float sA[64 * 130];  // 33280 B

  int Meff = Mcap;
  if (dM) { int m = *dM; Meff = m < Mcap ? m : Mcap; }
  long long rowBase = (long long)blockIdx.x * 64;
  if (rowBase >= Meff) return;

  // Stage W pair-packed: thread i -> (pair p, col n); both global reads coalesced,
  // LDS b64 stores conflict-free (banks 2n mod 64 distinct across a wave half).
  for (int i = threadIdx.x; i < 64 * HDIM; i += 128) {
    int p = i >> 7;
    int n = i & 127;
    v2f t;
    t.x = W[(2 * p) * HDIM + n];
    t.y = W[(2 * p + 1) * HDIM + n];
    *(v2f*)&sW[p * WP + n * 2] = t;
  }
  // Stage A tile (64x128), optional gather + pre-add, zero pad OOB rows
  for (int i = threadIdx.x; i < 64 * 32; i += 128) {
    int r = i >> 5, c = (i & 31) << 2;
    long long g = rowBase + r;
    float4 v = make_float4(0.0f, 0.0f, 0.0f, 0.0f);
    if (g < Meff) {
      long long sr = rowIdx ? (long long)rowIdx[g] : g;
      v = *(const float4*)(A + sr * HDIM + c);
      if (A2) {
        const float4 u = *(const float4*)(A2 + g * HDIM + c);
        v.x += u.x; v.y += u.y; v.z += u.z; v.w += u.w;
      }
    }
    float* d = &sA[r * 130 + c];
    d[0] = v.x; d[1] = v.y; d[2] = v.z; d[3] = v.w;
  }
  __syncthreads();

  const int wave = threadIdx.x >> 5;
  const int lane = threadIdx.x & 31;
  const int lm = lane & 15;   // M (for A) / N (for B,C)
  const int lh = lane >> 4;   // K-half selector

  v8f acc[8];
#pragma unroll
  for (int n = 0; n < 8; ++n) {
    float bv = bias ? bias[n * 16 + lm] : 0.0f;
#pragma unroll
    for (int r = 0; r < 8; ++r) acc[n][r] = bv;
  }

  const int am = wave * 16 + lm;
#pragma unroll 4
  for (int kb = 0; kb < HDIM; kb += 4) {
    const v2f a = *(const v2f*)&sA[am * 130 + kb + 2 * lh];
    const float* wrow = &sW[((kb >> 1) + lh) * WP + lm * 2];
#pragma unroll
    for (int n = 0; n < 8; ++n) {
      const v2f b = *(const v2f*)&wrow[n * 32];
      acc[n] = __builtin_amdgcn_wmma_f32_16x16x4_f32(
          false, a, false, b, (short)0, acc[n], false, false);
    }
  }

#pragma unroll
  for (int n = 0; n < 8; ++n) {
#pragma unroll
    for (int r = 0; r < 8; ++r) {
      long long gm = rowBase + wave * 16 + r + 8 * lh;
      if (gm < Meff) {
        float v = acc[n][r];
        if (doRelu) v = fmaxf(v, 0.0f);
        Out[gm * HDIM + n * 16 + lm] = v;
      }
    }
  }
}

// ---------------------------------------------------------------------------
// BatchNorm column stats: sums[c] += sum(X[:,c]); sums[128+c] += sum(X[:,c]^2)
// ---------------------------------------------------------------------------
__global__ void bn_stats(const float* __restrict__ X, float* sums,
                         int Mcap, const int* __restrict__ dM) {
  int M = Mcap;
  if (dM) { int m = *dM; M = m < Mcap ? m : Mcap; }
  int c = threadIdx.x;
  long long r0 = (long long)blockIdx.x * BN_ROWS;
  float s = 0.0f, q = 0.0f;
  for (int i = 0; i < BN_ROWS; ++i) {
    long long r = r0 + i;
    if (r < M) {
      float v = X[r * HDIM + c];
      s += v;
      q += v * v;
    }
  }
  unsafeAtomicAdd(&sums[c], s);
  unsafeAtomicAdd(&sums[HDIM + c], q);
}

__global__ void bn_fin(const float* __restrict__ sums, float* mu_istd,
                       int Mcap, const int* __restrict__ dM) {
  int M = Mcap;
  if (dM) { int m = *dM; M = m < Mcap ? m : Mcap; }
  int c = threadIdx.x;
  float inv = 1.0f / (float)M;
  float mu = sums[c] * inv;
  float var = sums[HDIM + c] * inv - mu * mu;
  mu_istd[c] = mu;
  mu_istd[HDIM + c] = rsqrtf(var + 1e-5f);
}

// ---------------------------------------------------------------------------
// Final: out = relu(skip + bn1(h1raw) + bn2(h2raw)[cid] + yvv[cid] + ykk[sb]) * valid
// ---------------------------------------------------------------------------
__global__ void final_combine(
    const float* __restrict__ skipO, const float* __restrict__ h1raw,
    const float* __restrict__ h2raw, const float* __restrict__ yvv,
    const float* __restrict__ ykk,
    const int* __restrict__ node_ids, const int* __restrict__ sub_batch,
    const unsigned char* __restrict__ valid,
    const float* __restrict__ bn1, const float* __restrict__ bn2,
    const float* __restrict__ g1, const float* __restrict__ be1,
    const float* __restrict__ g2, const float* __restrict__ be2,
    float* __restrict__ out, long long Nf) {
  long long t = (long long)blockIdx.x * blockDim.x + threadIdx.x;
  if (t >= Nf * 32) return;
  long long i = t >> 5;
  int c = (int)(t & 31) << 2;
  int nid = node_ids[i];
  long long cid = nid > 0 ? (long long)nid : 0;
  long long sb = sub_batch[i];
  float vf = valid[i] ? 1.0f : 0.0f;

  const float4 sk = *(const float4*)(skipO + i * HDIM + c);
  const float4 h1 = *(const float4*)(h1raw + i * HDIM + c);
  const float4 h2 = *(const float4*)(h2raw + cid * HDIM + c);
  const float4 vv = *(const float4*)(yvv + cid * HDIM + c);
  const float4 kk = *(const float4*)(ykk + sb * HDIM + c);

  float a1[4] = {h1.x, h1.y, h1.z, h1.w};
  float a2[4] = {h2.x, h2.y, h2.z, h2.w};
  float as[4] = {sk.x, sk.y, sk.z, sk.w};
  float av[4] = {vv.x, vv.y, vv.z, vv.w};
  float ak[4] = {kk.x, kk.y, kk.z, kk.w};
  float o[4];
#pragma unroll
  for (int j = 0; j < 4; ++j) {
    int cc = c + j;
    float t1 = (a1[j] - bn1[cc]) * bn1[HDIM + cc] * g1[cc] + be1[cc];
    float t2 = (a2[j] - bn2[cc]) * bn2[HDIM + cc] * g2[cc] + be2[cc];
    float s = as[j] + t1 + t2 + av[j] + ak[j];
    o[j] = fmaxf(s, 0.0f) * vf;
  }
  float4 r;
  r.x = o[0]; r.y = o[1]; r.z = o[2]; r.w = o[3];
  *(float4*)(out + i * HDIM + c) = r;
}

// ---------------------------------------------------------------------------
// Host launch
// ---------------------------------------------------------------------------
extern "C" void kernel_launch(void* const* d_in, const int* in_sizes, int n_in,
                              void* d_out, int out_size, void* d_ws, size_t ws_size,
                              hipStream_t stream) {
  const float* h_flat = (const float*)d_in[0];
  const int* intra_ei = (const int*)d_in[1];
  const float* ea_flat = (const float*)d_in[2];
  const unsigned char* valid = (const unsigned char*)d_in[3];
  const int* node_ids = (const int*)d_in[4];
  const int* dN = (const int*)d_in[5];  // N_total (device scalar)
  const int* edge_index = (const int*)d_in[6];
  const float* edge_attr = (const float*)d_in[7];
  const int* sub_batch = (const int*)d_in[8];
  const int* root_flat_idx = (const int*)d_in[11];
  const float* lw1 = (const float*)d_in[13];
  const float* lb1 = (const float*)d_in[14];
  const float* lw2 = (const float*)d_in[15];
  const float* lb2 = (const float*)d_in[16];
  const float* lgam = (const float*)d_in[17];
  const float* lbet = (const float*)d_in[18];
  const float* gw1 = (const float*)d_in[19];
  const float* gb1 = (const float*)d_in[20];
  const float* gw2 = (const float*)d_in[21];
  const float* gb2 = (const float*)d_in[22];
  const float* ggam = (const float*)d_in[23];
  const float* gbet = (const float*)d_in[24];
  const float* skw = (const float*)d_in[25];
  const float* skb = (const float*)d_in[26];
  const float* vvw = (const float*)d_in[27];
  const float* vvb = (const float*)d_in[28];
  const float* kkw = (const float*)d_in[29];
  const float* kkb = (const float*)d_in[30];

  const long long Nf = in_sizes[0] / HDIM;
  const long long Ei = in_sizes[1] / 2;
  const long long Eg = in_sizes[6] / 2;
  const long long S = in_sizes[11];

  // Workspace layout (floats)
  float* ws = (float*)d_ws;
  long long o = 0;
  float* bufA = ws + o; o += Nf * HDIM;                 // agg1 -> t1 -> h1raw
  float* bufB = ws + o; o += Nf * HDIM;                 // skip out
  float* bufC = ws + o; o += (long long)NT_CAP * HDIM;  // x_sum
  float* bufD = ws + o; o += (long long)NT_CAP * HDIM;  // agg2 -> t2 -> h2raw
  float* bufE = ws + o; o += (long long)NT_CAP * HDIM;  // vv_sum -> vvc -> yvv
  float* bufF = ws + o; o += S * HDIM;                  // ykk
  float* cntC = ws + o; o += NT_CAP;
  float* cntE = ws + o; o += NT_CAP;
  float* stat1 = ws + o; o += 2 * HDIM;
  float* stat2 = ws + o; o += 2 * HDIM;
  float* bn1 = ws + o; o += 2 * HDIM;
  float* bn2 = ws + o; o += 2 * HDIM;

  // Zero accumulation buffers (cntC..bn2 are contiguous)
  zero_f32<<<2048, 256, 0, stream>>>(bufA, Nf * HDIM);
  zero_f32<<<1024, 256, 0, stream>>>(bufC, (long long)NT_CAP * HDIM);
  zero_f32<<<1024, 256, 0, stream>>>(bufD, (long long)NT_CAP * HDIM);
  zero_f32<<<1024, 256, 0, stream>>>(bufE, (long long)NT_CAP * HDIM);
  zero_f32<<<128, 256, 0, stream>>>(cntC, 2LL * NT_CAP + 8 * HDIM);

  const dim3 blk(128);
  const int gNf = (int)((Nf + 63) / 64);
  const int gNt = (NT_CAP + 63) / 64;
  const int gS = (int)((S + 63) / 64);

  // h_skip = h_flat @ skip_w + skip_b
  gemm128_wmma<<<gNf, blk, 0, stream>>>(h_flat, nullptr, nullptr, skw, skb,
                                        bufB, (int)Nf, nullptr, 0);
  // intra-edge aggregation -> bufA
  {
    long long T = Ei * 32;
    edge_msg<<<(unsigned)((T + 255) / 256), 256, 0, stream>>>(h_flat, intra_ei,
                                                              ea_flat, bufA, Ei);
  }
  // x_sum scatter-mean numerators -> bufC, cntC
  {
    long long T = Nf * 32;
    scatter_rows<<<(unsigned)((T + 255) / 256), 256, 0, stream>>>(
        h_flat, node_ids, nullptr, bufC, cntC, Nf);
  }
  // vv canonical scatter-mean -> bufE, cntE (rows gathered via root_flat_idx)
  {
    long long T = S * 32;
    scatter_rows<<<(unsigned)((T + 255) / 256), 256, 0, stream>>>(
        h_flat, node_ids, root_flat_idx, bufE, cntE, S);
  }
  {
    long long T = (long long)NT_CAP * 32;
    div_rows<<<(unsigned)((T + 255) / 256), 256, 0, stream>>>(bufC, cntC, NT_CAP, dN);
    div_rows<<<(unsigned)((T + 255) / 256), 256, 0, stream>>>(bufE, cntE, NT_CAP, dN);
  }
  // global-edge aggregation over x_sum -> bufD
  {
    long long T = Eg * 32;
    edge_msg<<<(unsigned)((T + 255) / 256), 256, 0, stream>>>(bufC, edge_index,
                                                              edge_attr, bufD, Eg);
  }
  // local MLP: t1 = relu((h + agg1) @ w1 + b1); h1raw = t1 @ w2 + b2
  gemm128_wmma<<<gNf, blk, 0, stream>>>(h_flat, bufA, nullptr, lw1, lb1,
                                        bufA, (int)Nf, nullptr, 1);
  gemm128_wmma<<<gNf, blk, 0, stream>>>(bufA, nullptr, nullptr, lw2, lb2,
                                        bufA, (int)Nf, nullptr, 0);
  // global MLP on canonical nodes
  gemm128_wmma<<<gNt, blk, 0, stream>>>(bufC, bufD, nullptr, gw1, gb1,
                                        bufD, NT_CAP, dN, 1);
  gemm128_wmma<<<gNt, blk, 0, stream>>>(bufD, nullptr, nullptr, gw2, gb2,
                                        bufD, NT_CAP, dN, 0);
  // yvv = vvc @ vv_w + vv_b (gathered later in epilogue)
  gemm128_wmma<<<gNt, blk, 0, stream>>>(bufE, nullptr, nullptr, vvw, vvb,
                                        bufE, NT_CAP, dN, 0);
  // ykk = h_flat[root] @ kk_w + kk_b (gathered by sub_batch in epilogue)
  gemm128_wmma<<<gS, blk, 0, stream>>>(h_flat, nullptr, root_flat_idx, kkw, kkb,
                                       bufF, (int)S, nullptr, 0);
  // BatchNorm stats
  bn_stats<<<(int)((Nf + BN_ROWS - 1) / BN_ROWS), 128, 0, stream>>>(bufA, stat1,
                                                                    (int)Nf, nullptr);
  bn_stats<<<(NT_CAP + BN_ROWS - 1) / BN_ROWS, 128, 0, stream>>>(bufD, stat2,
                                                                 NT_CAP, dN);
  bn_fin<<<1, 128, 0, stream>>>(stat1, bn1, (int)Nf, nullptr);
  bn_fin<<<1, 128, 0, stream>>>(stat2, bn2, NT_CAP, dN);
  // Fused epilogue
  {
    long long T = Nf * 32;
    final_combine<<<(unsigned)((T + 255) / 256), 256, 0, stream>>>(
        bufB, bufA, bufD, bufE, bufF, node_ids, sub_batch, valid, bn1, bn2,
        lgam, lbet, ggam, gbet, (float*)d_out, Nf);
  }
}